// GraphAttention_32023276159482
// MI455X (gfx1250) — compile-verified
//
#include <hip/hip_runtime.h>

typedef __attribute__((ext_vector_type(2))) float v2f;
typedef __attribute__((ext_vector_type(4))) float v4f;
typedef __attribute__((ext_vector_type(8))) float v8f;

namespace {
constexpr int   KNN             = 20;
constexpr int   FOUT            = 16;
constexpr int   NPTS            = 1024;                // N
constexpr int   BATCH           = 512;                 // B
constexpr int   PTS             = BATCH * NPTS;        // 524288 points
constexpr int   PTS_PER_WAVE    = 4;
constexpr int   ROWS_PER_WAVE   = PTS_PER_WAVE * KNN;  // 80 rows
constexpr int   TILES_PER_WAVE  = ROWS_PER_WAVE / 16;  // 5 WMMA tiles
constexpr int   WAVES_PER_BLOCK = 8;                   // 256 threads
constexpr int   LDS_ROW         = 20;                  // padded row (16B-aligned, conflict-breaking)
constexpr int   PC_MAX          = PTS * 3 - 1;         // clamp for pad-lane loads
constexpr int   KNN_MAX         = PTS * KNN * 3 - 1;
constexpr float EPSV            = 0.001f;
}

__global__ __launch_bounds__(WAVES_PER_BLOCK * 32)
void ga_fused_kernel(
    const float* __restrict__ pc,   const float* __restrict__ knn,
    const float* __restrict__ w_s1, const float* __restrict__ b_s1,
    const float* __restrict__ g_s1, const float* __restrict__ be_s1,
    const float* __restrict__ m_s1, const float* __restrict__ v_s1,
    const float* __restrict__ w_s2, const float* __restrict__ b_s2,
    const float* __restrict__ g_s2, const float* __restrict__ be_s2,
    const float* __restrict__ m_s2, const float* __restrict__ v_s2,
    const float* __restrict__ w_n1, const float* __restrict__ b_n1,
    const float* __restrict__ g_n1, const float* __restrict__ be_n1,
    const float* __restrict__ m_n1, const float* __restrict__ v_n1,
    const float* __restrict__ w_n2, const float* __restrict__ b_n2,
    const float* __restrict__ g_n2, const float* __restrict__ be_n2,
    const float* __restrict__ m_n2, const float* __restrict__ v_n2,
    const float* __restrict__ out_bias,
    float* __restrict__ att_out, float* __restrict__ gf_out,
    float* __restrict__ coeff_out)
{
    // per-wave private LDS slabs -> no block barriers needed (DS is in-order per wave)
    __shared__ __align__(16) float gf_lds[WAVES_PER_BLOCK][ROWS_PER_WAVE][LDS_ROW]; // 50 KB
    __shared__ float cf_lds[WAVES_PER_BLOCK][ROWS_PER_WAVE];                        // 2.5 KB

    const int lane  = threadIdx.x & 31;
    const int wid   = threadIdx.x >> 5;
    const int gwave = blockIdx.x * WAVES_PER_BLOCK + wid;
    const int p0    = gwave * PTS_PER_WAVE;       // first point of this wave
    const int r0    = p0 * KNN;                   // first (b,n,k)-row

    const int m16 = lane & 15;     // A: row-in-tile / B,C,D: feature column N
    const int g   = lane >> 4;     // K-pair half (K = 2g, 2g+1)

    // ---- B matrix: W_n1 (3x16), K padded 3->4 with zeros (K = 2g + j per VGPR j) ----
    v2f bmat;
    bmat.x = w_n1[(2 * g) * FOUT + m16];                       // K = 0 or 2
    bmat.y = (g == 0) ? w_n1[1 * FOUT + m16] : 0.0f;           // K = 1 or pad

    // per-feature (N = m16) epilogue constants for dense1/BN1
    const float sc1 = g_n1[m16] * rsqrtf(v_n1[m16] + EPSV);
    const float sh1 = be_n1[m16] - m_n1[m16] * sc1;
    const float bi1 = b_n1[m16];
    // scalar BN2 constants
    const float sc2 = g_n2[0] * rsqrtf(v_n2[0] + EPSV);
    const float sh2 = be_n2[0] - m_n2[0] * sc2;
    const float bi2 = b_n2[0];
    // full W_n2 vector (uniform -> scalar regs)
    float w2v[FOUT];
#pragma unroll
    for (int f = 0; f < FOUT; ++f) w2v[f] = w_n2[f];

    // ================= Phase 1: 5 WMMA tiles (80 rows) =================
    for (int t = 0; t < TILES_PER_WAVE; ++t) {
        const int r  = r0 + t * 16 + m16;         // this lane's A row
        const int p  = r / KNN;                   // its point index
        const int ka = 2 * g;

        if (t + 1 < TILES_PER_WAVE)               // global_prefetch_b8 next tile
            __builtin_prefetch(&knn[(r + 16) * 3], 0, 0);

        // branchless diff loads (clamp keeps pad-lane addresses in bounds)
        const int ki0 = r * 3 + ka;
        const int ki1 = min(ki0 + 1, KNN_MAX);
        const int pi0 = p * 3 + ka;
        const int pi1 = min(pi0 + 1, PC_MAX);
        const float a0 = pc[pi0] - knn[ki0];
        const float a1 = pc[pi1] - knn[ki1];
        v2f amat;
        amat.x = a0;
        amat.y = (g == 0) ? a1 : 0.0f;            // K=3 pad via cndmask

        v8f c = {};
        c = __builtin_amdgcn_wmma_f32_16x16x4_f32(
                /*neg_a=*/false, amat, /*neg_b=*/false, bmat,
                /*c_mod=*/(short)0, c, /*reuse_a=*/false, /*reuse_b=*/false);

        // Epilogue: bias+relu+BN1 -> gf into LDS (C layout: M = v + 8g, N = m16)
#pragma unroll
        for (int v = 0; v < 8; ++v) {
            const float x = fmaxf(c[v] + bi1, 0.0f);
            gf_lds[wid][t * 16 + v + 8 * g][m16] = fmaf(x, sc1, sh1);
        }
        asm volatile("" ::: "memory");            // keep DS order (same-wave RAW is in-order)

        // Coalesced global store: lane (g,m16) -> row m16, features [8g, 8g+8)
        const int srow = t * 16 + m16;
        const v4f lo = *(const v4f*)&gf_lds[wid][srow][8 * g];
        const v4f hi = *(const v4f*)&gf_lds[wid][srow][8 * g + 4];
        float* dst = gf_out + (size_t)(r0 + srow) * FOUT + 8 * g;
        *(v4f*)(dst)     = lo;
        *(v4f*)(dst + 4) = hi;
    }

    asm volatile("" ::: "memory");

    // ---- sa MLP constants (feature = m16) ----
    const float scs1 = g_s1[m16] * rsqrtf(v_s1[m16] + EPSV);
    const float shs1 = be_s1[m16] - m_s1[m16] * scs1;
    const float bis1 = b_s1[m16];
    const float ws2f = w_s2[m16];
    const float scs2 = g_s2[0] * rsqrtf(v_s2[0] + EPSV);
    const float shs2 = be_s2[0] - m_s2[0] * scs2;
    const float bis2 = b_s2[0];

    // ---- sa for the 4 points, two pair-passes (half-wave per point) ----
    float sa[PTS_PER_WAVE];
#pragma unroll
    for (int j = 0; j < 2; ++j) {
        const int p = p0 + 2 * j + g;
        float h = bis1;
        h = fmaf(pc[p * 3 + 0], w_s1[0 * FOUT + m16], h);
        h = fmaf(pc[p * 3 + 1], w_s1[1 * FOUT + m16], h);
        h = fmaf(pc[p * 3 + 2], w_s1[2 * FOUT + m16], h);
        h = fmaf(fmaxf(h, 0.0f), scs1, shs1) * ws2f;
        h += __shfl_xor(h, 1, 32);
        h += __shfl_xor(h, 2, 32);
        h += __shfl_xor(h, 4, 32);
        h += __shfl_xor(h, 8, 32);
        const float s = fmaf(fmaxf(h + bis2, 0.0f), scs2, shs2);
        sa[2 * j + 0] = __shfl(s, 0, 32);
        sa[2 * j + 1] = __shfl(s, 16, 32);
    }

    // ================= Phase 2: per point (na + softmax + att) =================
#pragma unroll
    for (int i = 0; i < PTS_PER_WAVE; ++i) {
        const int p  = p0 + i;
        const int nc = p & (NPTS - 1);            // n within cloud (N=1024)
        const bool valid = lane < KNN;
        const int row = i * KNN + (valid ? lane : 0);

        // ---- na: in-lane full-row dot gf[row] . w_n2 (lands in softmax lane) ----
        const v4f q0 = *(const v4f*)&gf_lds[wid][row][0];
        const v4f q1 = *(const v4f*)&gf_lds[wid][row][4];
        const v4f q2 = *(const v4f*)&gf_lds[wid][row][8];
        const v4f q3 = *(const v4f*)&gf_lds[wid][row][12];
        float dot = 0.0f;
#pragma unroll
        for (int f = 0; f < 4; ++f) {
            dot = fmaf(q0[f], w2v[f],      dot);
            dot = fmaf(q1[f], w2v[4 + f],  dot);
            dot = fmaf(q2[f], w2v[8 + f],  dot);
            dot = fmaf(q3[f], w2v[12 + f], dot);
        }
        const float na = fmaf(fmaxf(dot + bi2, 0.0f), sc2, sh2);

        // ---- leaky-relu + softmax over K=20 (lanes 0..19) ----
        float lg = sa[i] + na;
        lg = (lg > 0.0f) ? lg : 0.2f * lg;

        float mx = valid ? lg : -3.402823466e38f;
        mx = fmaxf(mx, __shfl_xor(mx,  1, 32));
        mx = fmaxf(mx, __shfl_xor(mx,  2, 32));
        mx = fmaxf(mx, __shfl_xor(mx,  4, 32));
        mx = fmaxf(mx, __shfl_xor(mx,  8, 32));
        mx = fmaxf(mx, __shfl_xor(mx, 16, 32));

        const float e = valid ? expf(lg - mx) : 0.0f;
        float sum = e;
        sum += __shfl_xor(sum,  1, 32);
        sum += __shfl_xor(sum,  2, 32);
        sum += __shfl_xor(sum,  4, 32);
        sum += __shfl_xor(sum,  8, 32);
        sum += __shfl_xor(sum, 16, 32);

        const float cf = e / sum;
        if (valid) {
            coeff_out[(size_t)p * KNN + lane] = cf;
            cf_lds[wid][i * KNN + lane] = cf;
        }
        asm volatile("" ::: "memory");

        // ---- att[f] = sum_k coeff[k]*gf[k][f] + out_bias; halves split k ----
        float acc = 0.0f;
#pragma unroll
        for (int kk = 0; kk < KNN / 2; ++kk) {
            const int k = g * (KNN / 2) + kk;
            acc = fmaf(cf_lds[wid][i * KNN + k], gf_lds[wid][i * KNN + k][m16], acc);
        }
        acc += __shfl_xor(acc, 16, 32);
        const float att = fmaxf(acc + out_bias[nc * FOUT + m16], 0.0f);
        if (g == 0)
            att_out[(size_t)p * FOUT + m16] = att;
    }
}

extern "C" void kernel_launch(void* const* d_in, const int* in_sizes, int n_in,
                              void* d_out, int out_size, void* d_ws, size_t ws_size,
                              hipStream_t stream) {
    (void)in_sizes; (void)n_in; (void)out_size; (void)d_ws; (void)ws_size;

    const float* pc    = (const float*)d_in[0];
    const float* knn   = (const float*)d_in[1];
    const float* w_s1  = (const float*)d_in[2];
    const float* b_s1  = (const float*)d_in[3];
    const float* g_s1  = (const float*)d_in[4];
    const float* be_s1 = (const float*)d_in[5];
    const float* m_s1  = (const float*)d_in[6];
    const float* v_s1  = (const float*)d_in[7];
    const float* w_s2  = (const float*)d_in[8];
    const float* b_s2  = (const float*)d_in[9];
    const float* g_s2  = (const float*)d_in[10];
    const float* be_s2 = (const float*)d_in[11];
    const float* m_s2  = (const float*)d_in[12];
    const float* v_s2  = (const float*)d_in[13];
    const float* w_n1  = (const float*)d_in[14];
    const float* b_n1  = (const float*)d_in[15];
    const float* g_n1  = (const float*)d_in[16];
    const float* be_n1 = (const float*)d_in[17];
    const float* m_n1  = (const float*)d_in[18];
    const float* v_n1  = (const float*)d_in[19];
    const float* w_n2  = (const float*)d_in[20];
    const float* b_n2  = (const float*)d_in[21];
    const float* g_n2  = (const float*)d_in[22];
    const float* be_n2 = (const float*)d_in[23];
    const float* m_n2  = (const float*)d_in[24];
    const float* v_n2  = (const float*)d_in[25];
    const float* obia  = (const float*)d_in[26];

    // outputs concatenated flat: att (B*N*16), graph_features (B*N*20*16), coeff (B*N*20)
    float* att = (float*)d_out;
    float* gf  = att + (size_t)PTS * FOUT;
    float* cf  = gf  + (size_t)PTS * KNN * FOUT;

    const int blocks = PTS / (PTS_PER_WAVE * WAVES_PER_BLOCK);   // 16384
    ga_fused_kernel<<<blocks, WAVES_PER_BLOCK * 32, 0, stream>>>(
        pc, knn,
        w_s1, b_s1, g_s1, be_s1, m_s1, v_s1,
        w_s2, b_s2, g_s2, be_s2, m_s2, v_s2,
        w_n1, b_n1, g_n1, be_n1, m_n1, v_n1,
        w_n2, b_n2, g_n2, be_n2, m_n2, v_n2,
        obia, att, gf, cf);
}